// DynamicCostVolume_39247411151068
// MI455X (gfx1250) — compile-verified
//
#include <hip/hip_runtime.h>

// Problem constants (from setup_inputs: B=4, C=64, H=128, W=256, D=48+1, groups=8)
#define B_   4
#define C_   64
#define H_   128
#define W_   256
#define G_   8
#define CPG_ 8
#define D_   49

// Raw vector types (HIP's float4 is a class; builtins need real vectors).
typedef float v4f __attribute__((vector_size(16)));
typedef int   v4i __attribute__((vector_size(16)));

#if defined(__HIP_DEVICE_COMPILE__) && __has_builtin(__builtin_amdgcn_global_load_async_to_lds_b128)
#define USE_ASYNC_LDS 1
#else
#define USE_ASYNC_LDS 0
#endif
#if defined(__HIP_DEVICE_COMPILE__) && __has_builtin(__builtin_amdgcn_s_wait_asynccnt)
#define HAVE_WAIT_ASYNC 1
#else
#define HAVE_WAIT_ASYNC 0
#endif

__global__ __launch_bounds__(256) void
cost_volume_kernel(const float* __restrict__ left,
                   const float* __restrict__ right,
                   float* __restrict__ out)
{
    // Stage the full `right` feature row for this (b, h): 64 ch x 256 w fp32 = 64 KB LDS.
    __shared__ __align__(16) float Rsh[C_ * W_];

    const int h   = blockIdx.x;   // 0..127
    const int b   = blockIdx.y;   // 0..3
    const int tid = threadIdx.x;  // 0..255

    // ---------------- Phase 1: right row -> LDS (async b128 copies, ASYNCcnt) ----------------
    // 4096 float4 chunks total, 16 per thread. LDS layout: Rsh[ch*W + w].
#pragma unroll
    for (int k = 0; k < 16; ++k) {
        const int t  = k * 256 + tid;   // 16-byte chunk id
        const int ch = t >> 6;          // 64 chunks per channel
        const int p4 = t & 63;
        const float* gp = right + ((size_t)(b * C_ + ch) * H_ + h) * W_ + (p4 << 2);
#if USE_ASYNC_LDS
        __builtin_amdgcn_global_load_async_to_lds_b128(
            (__attribute__((address_space(1))) v4i*)gp,
            (__attribute__((address_space(3))) v4i*)(&Rsh[t << 2]),
            0, 0);
#else
        *(v4f*)(&Rsh[t << 2]) = *(const v4f*)gp;
#endif
    }
#if USE_ASYNC_LDS
#if HAVE_WAIT_ASYNC
    __builtin_amdgcn_s_wait_asynccnt(0);
#else
    asm volatile("s_wait_asynccnt 0x0" ::: "memory");
#endif
#endif
    __syncthreads();

    // ---------------- Phase 2: sliding-window correlation over disparities ----------------
    // Thread owns 4 consecutive w (w0..w0+3) and two channels c = cq and cq+4.
    const int wq = tid & 63;
    const int w0 = wq << 2;
    const int cq = tid >> 6;  // 0..3

#pragma unroll
    for (int cc = 0; cc < 2; ++cc) {
        const int c = cq + 4 * cc;  // 0..7

        // Left operand, pre-scaled by 1/groups, read-once coalesced b128 loads.
        v4f l4[G_];
#pragma unroll
        for (int g = 0; g < G_; ++g) {
            const int ch = g * CPG_ + c;
            v4f v = *(const v4f*)(left + ((size_t)(b * C_ + ch) * H_ + h) * W_ + w0);
            l4[g] = v * 0.125f;
        }

        // Register window rw[g][j] = R[ch][w0 + j - d]; init at d = 0 (ds_load_b128).
        float rw[G_][4];
#pragma unroll
        for (int g = 0; g < G_; ++g) {
            v4f v = *(const v4f*)(&Rsh[(g * CPG_ + c) * W_ + w0]);
            rw[g][0] = v[0]; rw[g][1] = v[1]; rw[g][2] = v[2]; rw[g][3] = v[3];
        }

        float* outc = out + ((size_t)(b * CPG_ + c) * D_ * H_ + h) * W_;

#pragma unroll 7
        for (int d = 0; d < D_; ++d) {
            v4f acc = {0.f, 0.f, 0.f, 0.f};
#pragma unroll
            for (int g = 0; g < G_; ++g) {
                acc[0] = fmaf(l4[g][0], rw[g][0], acc[0]);
                acc[1] = fmaf(l4[g][1], rw[g][1], acc[1]);
                acc[2] = fmaf(l4[g][2], rw[g][2], acc[2]);
                acc[3] = fmaf(l4[g][3], rw[g][3], acc[3]);
            }
            // Streaming (nontemporal) coalesced b128 store: write-once 205 MB stream,
            // keeps the 192 MB L2 free for the (reused) inputs.
            __builtin_nontemporal_store(acc, (v4f*)(outc + (size_t)d * H_ * W_ + w0));

            // Shift window for d+1: one new LDS dword per group, zero-guarded at w<d
            // (this also implements the reference's zero padding for w < d).
            const int idx  = w0 - d - 1;
            const int idxc = idx < 0 ? 0 : idx;
#pragma unroll
            for (int g = 0; g < G_; ++g) {
                rw[g][3] = rw[g][2];
                rw[g][2] = rw[g][1];
                rw[g][1] = rw[g][0];
                const float v = Rsh[(g * CPG_ + c) * W_ + idxc];
                rw[g][0] = (idx < 0) ? 0.0f : v;
            }
        }
    }
}

extern "C" void kernel_launch(void* const* d_in, const int* in_sizes, int n_in,
                              void* d_out, int out_size, void* d_ws, size_t ws_size,
                              hipStream_t stream) {
    (void)in_sizes; (void)n_in; (void)d_ws; (void)ws_size; (void)out_size;
    const float* left  = (const float*)d_in[0];
    const float* right = (const float*)d_in[1];
    float* out = (float*)d_out;

    dim3 grid(H_, B_);   // one workgroup per (h, b) row
    dim3 block(256);     // 8 wave32
    cost_volume_kernel<<<grid, block, 0, stream>>>(left, right, out);
}